// YOLOLoss_53738630808115
// MI455X (gfx1250) — compile-verified
//
#include <hip/hip_runtime.h>
#include <math.h>

// ---------------- problem constants ----------------
#define NCELL0 307200   // 16*3*80*80
#define NCELL1 76800    // 16*3*40*40
#define NCELL2 19200    // 16*3*20*20
#define OFF1   307200
#define OFF2   384000
#define NCELLS 403200
#define NBLK   256      // loss-kernel grid = partial count (16x16 matrix)

typedef float v2f __attribute__((ext_vector_type(2)));
typedef float v8f __attribute__((ext_vector_type(8)));

__device__ __forceinline__ float bce_f(float l, float t) {
    // max(l,0) - l*t + log1p(exp(-|l|))
    return fmaxf(l, 0.0f) - l * t + log1pf(expf(-fabsf(l)));
}

// ---------------- kernel 0: zero-init workspace & output ----------------
__global__ void init_kernel(float* __restrict__ flags, float* __restrict__ partials,
                            float* __restrict__ out) {
    int idx = blockIdx.x * blockDim.x + threadIdx.x;
    int stride = gridDim.x * blockDim.x;
    for (int g = idx; g < NCELLS; g += stride) flags[g] = 0.0f;
    if (idx < NBLK) partials[idx] = 0.0f;
    if (idx == 0) out[0] = 0.0f;
}

// ---------------- kernel 1: build compact targets ----------------
// One block per batch index b; serial over m to preserve the reference's
// sequential-scan overwrite semantics (cross-b writes never collide).
__global__ void build_targets_kernel(const float* __restrict__ targets,
                                     const float* __restrict__ anchors,
                                     float* __restrict__ flags,
                                     float* __restrict__ rec) {
    if (threadIdx.x != 0) return;
    const int b = blockIdx.x;                 // 0..15
    const float EPSf = 1e-16f;
    const float strideV[3] = {8.0f, 16.0f, 32.0f};
    const int   gsV[3]     = {80, 40, 20};
    const int   baseV[3]   = {0, OFF1, OFF2};
    for (int m = 0; m < 40; ++m) {
        const float* t = targets + (b * 40 + m) * 5;
        float c = t[0];
        float x = t[1] * 640.0f, y = t[2] * 640.0f;
        float w = t[3] * 640.0f, h = t[4] * 640.0f;
        float ps_max[3]; int ps_arg[3];
        for (int s = 0; s < 3; ++s) {
            float bm = -1.0f; int ba = 0;
            for (int a = 0; a < 3; ++a) {
                float aw = anchors[(s * 3 + a) * 2 + 0];
                float ah = anchors[(s * 3 + a) * 2 + 1];
                float inter = fminf(w, aw) * fminf(h, ah);
                float iou = inter / (w * h + aw * ah - inter + EPSf);
                if (iou > bm) { bm = iou; ba = a; }   // first-max (argmax semantics)
            }
            ps_max[s] = bm; ps_arg[s] = ba;
        }
        int best_scale = 0; float best_iou = ps_max[0];
        if (ps_max[1] > best_iou) { best_iou = ps_max[1]; best_scale = 1; }
        if (ps_max[2] > best_iou) { best_iou = ps_max[2]; best_scale = 2; }
        int best_anchor = ps_arg[best_scale];
        bool valid = best_iou >= 0.3f;
        for (int s = 0; s < 3; ++s) {
            float sv = strideV[s]; int gs = gsV[s];
            int i = (int)floorf(x / sv);
            int j = (int)floorf(y / sv);
            bool inb = (i >= 0) && (i < gs) && (j >= 0) && (j < gs);
            int ic = min(max(i, 0), gs - 1);
            int jc = min(max(j, 0), gs - 1);
            if (valid && (best_scale == s) && inb) {
                // NOTE: reference swaps i/j in the cell-offset computation; replicate.
                float x_cell = x / sv - (float)j;
                float y_cell = y / sv - (float)i;
                float aw = anchors[(s * 3 + best_anchor) * 2 + 0];
                float ah = anchors[(s * 3 + best_anchor) * 2 + 1];
                float w_cell = logf(w / aw + EPSf);
                float h_cell = logf(h / ah + EPSf);
                int cell = baseV[s] + (((b * 3 + best_anchor) * gs + jc) * gs + ic);
                flags[cell] = 1.0f;
                float* r = rec + (long)cell * 5;
                r[0] = x_cell; r[1] = y_cell; r[2] = w_cell; r[3] = h_cell; r[4] = c;
            }
            if (valid && (best_scale != s) && (ps_max[s] > 0.5f) && inb) {
                for (int a = 0; a < 3; ++a) {
                    int cell = baseV[s] + (((b * 3 + a) * gs + jc) * gs + ic);
                    flags[cell] = -1.0f;            // ignore marker on all anchors
                }
            }
        }
    }
}

// ---------------- kernel 2: fused loss over all cells ----------------
__global__ void loss_kernel(const float* __restrict__ p0, const float* __restrict__ p1,
                            const float* __restrict__ p2, const float* __restrict__ anchors,
                            const float* __restrict__ flags, const float* __restrict__ rec,
                            float* __restrict__ partials) {
    const int tid = blockIdx.x * blockDim.x + threadIdx.x;
    const int nstride = gridDim.x * blockDim.x;
    float acc = 0.0f;
    for (int g = tid; g < NCELLS; g += nstride) {
        int pg = g + nstride;
        if (pg < NCELLS) __builtin_prefetch(&flags[pg], 0, 1);   // global_prefetch_b8
        float flag = flags[g];
        if (flag == -1.0f) continue;                              // ignored cell
        int s, local, gs; const float* pred;
        if (g < OFF1)      { s = 0; local = g;        gs = 80; pred = p0; }
        else if (g < OFF2) { s = 1; local = g - OFF1; gs = 40; pred = p1; }
        else               { s = 2; local = g - OFF2; gs = 20; pred = p2; }
        const float* pc = pred + (long)local * 85;
        if (flag == 0.0f) {
            // noobj: only channel 4 needed
            acc += 0.1f * bce_f(pc[4], 0.0f);                     // LAMBDA_NOOBJ
            continue;
        }
        // obj cell (rare): full CIoU + obj + cls
        int i  = local % gs; int t1 = local / gs;
        int j  = t1 % gs;    int t2 = t1 / gs;
        int a  = t2 % 3;
        float aw = anchors[(s * 3 + a) * 2 + 0];
        float ah = anchors[(s * 3 + a) * 2 + 1];
        float balance = (s == 0) ? 4.0f : ((s == 1) ? 1.0f : 0.4f);
        const float* r = rec + (long)g * 5;
        float Wf = (float)gs;
        float px = 1.0f / (1.0f + expf(-pc[0]));
        float py = 1.0f / (1.0f + expf(-pc[1]));
        float b1x = (px + (float)i) / Wf, b1y = (py + (float)j) / Wf;
        float b1w = expf(pc[2]) * aw,     b1h = expf(pc[3]) * ah;
        float b2x = (r[0] + (float)i) / Wf, b2y = (r[1] + (float)j) / Wf;
        float b2w = expf(r[2]) * aw,        b2h = expf(r[3]) * ah;
        float b1x1 = b1x - b1w * 0.5f, b1x2 = b1x + b1w * 0.5f;
        float b1y1 = b1y - b1h * 0.5f, b1y2 = b1y + b1h * 0.5f;
        float b2x1 = b2x - b2w * 0.5f, b2x2 = b2x + b2w * 0.5f;
        float b2y1 = b2y - b2h * 0.5f, b2y2 = b2y + b2h * 0.5f;
        float iw = fmaxf(fminf(b1x2, b2x2) - fmaxf(b1x1, b2x1), 0.0f);
        float ih = fmaxf(fminf(b1y2, b2y2) - fmaxf(b1y1, b2y1), 0.0f);
        float inter = iw * ih;
        float w1 = b1x2 - b1x1, h1 = b1y2 - b1y1;
        float w2 = b2x2 - b2x1, h2 = b2y2 - b2y1;
        float uni = w1 * h1 + 1e-16f + w2 * h2 - inter;
        float iou = inter / uni;
        float cdx = b2x1 + b2x2 - b1x1 - b1x2;
        float cdy = b2y1 + b2y2 - b1y1 - b1y2;
        float cd  = cdx * cdx + cdy * cdy;
        float dd  = (b2x2 - b2x1) * (b2x2 - b2x1) + (b2y2 - b2y1) * (b2y2 - b2y1) + 1e-16f;
        float u   = cd / dd;
        float at  = atanf(w2 / (h2 + 1e-16f)) - atanf(w1 / (h1 + 1e-16f));
        float v   = 0.40528473456935109f * at * at;   // 4/pi^2
        float alpha = v / (1.0f - iou + v + 1e-16f);
        float ciou  = iou - u - alpha * v;
        acc += (-ciou) * 0.05f * balance;                          // LAMBDA_COORD
        float iou_pos = fmaxf(iou, 0.0f);
        acc += bce_f(pc[4], iou_pos) * balance;                    // LAMBDA_OBJ
        float cls = r[4];                                          // target ch5 = class id float
        for (int k = 0; k < 80; ++k) {
            float tk = (k == 0) ? cls : 0.0f;
            acc += 0.5f * bce_f(pc[5 + k], tk);                    // LAMBDA_CLS
        }
    }
    // wave32 reduction
    for (int m = 16; m >= 1; m >>= 1) acc += __shfl_xor(acc, m, 32);
    __shared__ float lds[8];
    int lane = threadIdx.x & 31, wave = threadIdx.x >> 5;
    if (lane == 0) lds[wave] = acc;
    __syncthreads();
    if (wave == 0) {
        float v2 = (lane < (int)(blockDim.x >> 5)) ? lds[lane] : 0.0f;
        for (int m = 4; m >= 1; m >>= 1) v2 += __shfl_xor(v2, m, 32);
        if (lane == 0) partials[blockIdx.x] = v2;
    }
}

// ---------------- kernel 3: WMMA finisher over 256 partials ----------------
// Treat partials as 16x16 f32 matrix M. D = ones(16x4) x B_chunk(4x16) + C,
// accumulated over 4 chunks => every row of D = column sums of M.
// B(4x16 f32) layout: VGPR0 = {K0 lanes0-15, K1 lanes16-31} -> p[64t+L];
//                     VGPR1 = {K2, K3}                      -> p[64t+32+L].
__global__ void final_reduce_kernel(const float* __restrict__ partials,
                                    float* __restrict__ out) {
    int lane = threadIdx.x;   // 32 threads, EXEC all-1s for WMMA
    v2f A; A[0] = 1.0f; A[1] = 1.0f;       // ones matrix (layout-invariant)
    v8f C = {};
    for (int t = 0; t < 4; ++t) {
        v2f B;
        B[0] = partials[t * 64 + lane];
        B[1] = partials[t * 64 + 32 + lane];
        C = __builtin_amdgcn_wmma_f32_16x16x4_f32(
            /*neg_a=*/false, A, /*neg_b=*/false, B,
            /*c_mod=*/(short)0, C, /*reuse_a=*/false, /*reuse_b=*/false);
    }
    float col = C[0];   // lanes hold colsum(lane & 15)
    col += __shfl_xor(col, 8, 32);
    col += __shfl_xor(col, 4, 32);
    col += __shfl_xor(col, 2, 32);
    col += __shfl_xor(col, 1, 32);
    if (lane == 0) out[0] = col;
}

extern "C" void kernel_launch(void* const* d_in, const int* in_sizes, int n_in,
                              void* d_out, int out_size, void* d_ws, size_t ws_size,
                              hipStream_t stream) {
    const float* p0      = (const float*)d_in[0];
    const float* p1      = (const float*)d_in[1];
    const float* p2      = (const float*)d_in[2];
    const float* targets = (const float*)d_in[3];
    const float* anchors = (const float*)d_in[4];
    float* out = (float*)d_out;

    // workspace layout: flags[NCELLS] | rec[NCELLS*5] | partials[NBLK]  (~9.7 MB)
    float* flags    = (float*)d_ws;
    float* rec      = flags + NCELLS;
    float* partials = rec + (size_t)NCELLS * 5;

    init_kernel<<<512, 256, 0, stream>>>(flags, partials, out);
    build_targets_kernel<<<16, 32, 0, stream>>>(targets, anchors, flags, rec);
    loss_kernel<<<NBLK, 256, 0, stream>>>(p0, p1, p2, anchors, flags, rec, partials);
    final_reduce_kernel<<<1, 32, 0, stream>>>(partials, out);
}